// WKNN_44856638439627
// MI455X (gfx1250) — compile-verified
//
#include <hip/hip_runtime.h>

// ---------------------------------------------------------------------------
// Weighted KNN (Q=1024, N=200000, d=64, K=32, 10 classes) for gfx1250 (CDNA5)
//  - distance GEMM via V_WMMA_F32_16X16X4_F32 (exact fp32 ordering)
//  - train chunks streamed with GLOBAL_LOAD_ASYNC_TO_LDS_B128, double-buffered,
//    synchronized with s_wait_asynccnt + workgroup barrier
//  - selection on distance keys (no division in hot path); exact IEEE 1/key
//    applied only to the 32 survivors per query
// ---------------------------------------------------------------------------

typedef __attribute__((ext_vector_type(2))) float v2f;
typedef __attribute__((ext_vector_type(4))) float v4f;
typedef __attribute__((ext_vector_type(8))) float v8f;
typedef __attribute__((ext_vector_type(4))) int   v4i;

// address-space-qualified pointer types for the async-copy builtin
typedef __attribute__((address_space(1))) v4i* gv4i_p;   // global
typedef __attribute__((address_space(3))) v4i* lv4i_p;   // LDS

#define QTOT    1024
#define DIM     64
#define XDIM    (DIM + 4)       // padded LDS row stride: conflict-free B-frag reads
#define NTRAIN  200000
#define KSEL    32
#define NCLS    10
#define NSPLIT  8               // train-set split across blocks for parallelism
#define CHUNK   128             // train rows per pipeline stage
#define SWS     (CHUNK + 33)    // padded key-tile stride (odd: kills 2-way conflicts)
#define TB      256             // 8 wave32 waves
#define MT      16              // queries per block (one WMMA M tile)
#define FINF    1e30f           // sentinel key (always loses a min-selection)

#if __has_builtin(__builtin_amdgcn_global_load_async_to_lds_b128)
#define KNN_HAS_ASYNC 1
#else
#define KNN_HAS_ASYNC 0
#endif

static __device__ __forceinline__ void knn_wait_async0() {
#if KNN_HAS_ASYNC
#if __has_builtin(__builtin_amdgcn_s_wait_asynccnt)
    __builtin_amdgcn_s_wait_asynccnt(0);
#else
    asm volatile("s_wait_asynccnt 0x0" ::: "memory");
#endif
#endif
}

// async 16B global -> LDS copy (per-lane addresses)
static __device__ __forceinline__ void knn_async_cp16(const float* src, float* dst) {
#if KNN_HAS_ASYNC
    float* s = const_cast<float*>(src);
    __builtin_amdgcn_global_load_async_to_lds_b128((gv4i_p)s, (lv4i_p)dst,
                                                   /*imm offset*/ 0, /*cpol*/ 0);
#else
    *(v4f*)dst = *(const v4f*)src;
#endif
}

// --------------------------- kernel 0: ||x||^2 ------------------------------
__global__ __launch_bounds__(256)
void knn_x2_kernel(const float* __restrict__ train, float* __restrict__ x2, int n) {
    int i = blockIdx.x * blockDim.x + threadIdx.x;
    if (i >= n) return;
    const float* r = train + (size_t)i * DIM;
    float s = 0.f;
#pragma unroll
    for (int j = 0; j < DIM; j += 4) {
        v4f v = *(const v4f*)(r + j);
        s += v.x * v.x + v.y * v.y + v.z * v.z + v.w * v.w;
    }
    x2[i] = s;
}

// ------------- kernel 1: WMMA distance keys + partial bottom-K --------------
__global__ __launch_bounds__(TB)
void knn_partial_kernel(const float* __restrict__ queries,
                        const float* __restrict__ train,
                        const float* __restrict__ x2,
                        float* __restrict__ wsW,
                        int*   __restrict__ wsI) {
    __shared__ float sQ[MT][DIM];           //  4 KB  query tile
    __shared__ float sQ2[MT];
    __shared__ float sX[2][CHUNK][XDIM];    // 68 KB  double-buffered train chunk
    __shared__ float sX2[2][CHUNK];         //  1 KB
    __shared__ float sW[MT][SWS];           // 10 KB  key tile staging
    __shared__ float sLw[TB][KSEL];         // 32 KB  per-thread partial keys
    __shared__ int   sLi[TB][KSEL];         // 32 KB  per-thread partial indices
    __shared__ int   sFi[MT][KSEL];         //  2 KB  merged indices

    const int t     = threadIdx.x;
    const int lane  = t & 31;
    const int wave  = t >> 5;
    const int qBase = blockIdx.x * MT;
    const int split = blockIdx.y;
    const int nBegin = split * (NTRAIN / NSPLIT);
    const int nEnd   = nBegin + (NTRAIN / NSPLIT);
    const int steps  = (nEnd - nBegin + CHUNK - 1) / CHUNK;   // 196

    // async (or fallback) stage of one train chunk into LDS buffer `buf`
    auto stage = [&](int chunkIdx, int buf) {
        const int nB = nBegin + chunkIdx * CHUNK;
#pragma unroll
        for (int i = 0; i < 8; ++i) {
            int f = i * TB + t;               // 0..2047
            int r = f >> 4;
            int c = (f & 15) << 2;
            int n = nB + r;
            if (n >= nEnd) n = nEnd - 1;      // clamp; masked via FINF key later
            knn_async_cp16(train + (size_t)n * DIM + c, &sX[buf][r][c]);
        }
        if (t < CHUNK) {
            int n = nB + t;
            sX2[buf][t] = (n < nEnd) ? x2[n] : 0.f;
            // prefetch hint two chunks ahead -> global_prefetch_b8
            int pn = nB + 2 * CHUNK + t;
            if (pn >= nEnd) pn = nEnd - 1;
            __builtin_prefetch(train + (size_t)pn * DIM, 0, 1);
        }
    };

    // ---- stage query tile (16x64 floats, coalesced b128) ----
    {
        v4f v = *(const v4f*)(queries + (size_t)qBase * DIM + t * 4);
        int m = t >> 4, c = (t & 15) << 2;
        *(v4f*)(&sQ[m][c]) = v;
    }
    // init per-thread partial lists (private rows, no race)
#pragma unroll
    for (int k = 0; k < KSEL; ++k) { sLw[t][k] = FINF; sLi[t][k] = 0; }
    __syncthreads();
    if (t < MT) {
        float s = 0.f;
#pragma unroll
        for (int j = 0; j < DIM; ++j) s += sQ[t][j] * sQ[t][j];
        sQ2[t] = s;
    }
    __syncthreads();

    // ---- preload all 16 A fragments (16x4 f32 tiles of the query matrix) ----
    // lane<16: M=lane, k-offsets {0,1}; lane>=16: M=lane-16, k-offsets {2,3}
    float ax[16], ay[16];
    {
        const int m = lane & 15, hi = lane >> 4;
#pragma unroll
        for (int kk = 0; kk < 16; ++kk) {
            ax[kk] = sQ[m][kk * 4 + 2 * hi];
            ay[kk] = sQ[m][kk * 4 + 2 * hi + 1];
        }
    }

    float curMax = FINF;     // running max (worst) key of this thread's list
    int   curMaxPos = 0;

    stage(0, 0);             // prologue: fill buffer 0

    for (int i = 0; i < steps; ++i) {
        const int cb = i & 1;
        const int nBase = nBegin + i * CHUNK;

        knn_wait_async0();   // this wave's async writes into buf[cb] done
        __syncthreads();     // all waves: buf[cb] ready; sW + buf[cb^1] free

        if (i + 1 < steps) stage(i + 1, cb ^ 1);   // overlap next fetch

        // ---- WMMA: wave computes its 16-column tile, K=64 in 16 steps ----
        {
            v8f acc = {0.f, 0.f, 0.f, 0.f, 0.f, 0.f, 0.f, 0.f};
            const int nl = (wave << 4) + (lane & 15);
            const int hi = lane >> 4;
#pragma unroll
            for (int kk = 0; kk < 16; ++kk) {
                v2f a; a.x = ax[kk]; a.y = ay[kk];
                v2f b = *(const v2f*)(&sX[cb][nl][kk * 4 + 2 * hi]);
                acc = __builtin_amdgcn_wmma_f32_16x16x4_f32(
                        false, a, false, b, (short)0, acc, false, false);
            }
            // distance keys (selection-equivalent to inverse-distance weights)
            const float xn    = sX2[cb][nl];
            const bool  valid = (nBase + nl) < nEnd;
#pragma unroll
            for (int r = 0; r < 8; ++r) {
                int   m = r + (hi << 3);
                float d = sQ2[m] + xn - 2.0f * acc[r];
                d = fmaxf(d, 0.0f);
                float key = (d == 0.0f) ? 0.1f : d;   // zero-distance rule
                sW[m][nl] = valid ? key : FINF;
            }
        }
        __syncthreads();

        // ---- selection: 16 threads per query keep the 32 smallest keys ----
        {
            const int m = t >> 4, j = t & 15;
#pragma unroll
            for (int c = 0; c < CHUNK / 16; ++c) {
                int   nl  = j + (c << 4);
                float key = sW[m][nl];
                if (key < curMax) {
                    sLw[t][curMaxPos] = key;
                    sLi[t][curMaxPos] = nBase + nl;
                    float mx = sLw[t][0]; int mp = 0;
#pragma unroll
                    for (int k = 1; k < KSEL; ++k) {
                        float v = sLw[t][k];
                        if (v > mx) { mx = v; mp = k; }
                    }
                    curMax = mx; curMaxPos = mp;
                }
            }
        }
    }
    __syncthreads();

    // ---- per-block merge: 16x32 partial lists -> 32 smallest per query ----
    if (t < MT) {
        const int m = t;
        float mx = FINF; int mp = 0;
#pragma unroll
        for (int k = 0; k < KSEL; ++k) { sW[m][k] = FINF; sFi[m][k] = 0; }
        for (int src = 0; src < 16; ++src) {
            int row = (m << 4) + src;
            for (int k = 0; k < KSEL; ++k) {
                float key = sLw[row][k];
                if (key < mx) {
                    sW[m][mp]  = key;
                    sFi[m][mp] = sLi[row][k];
                    float nmx = sW[m][0]; int nmp = 0;
#pragma unroll
                    for (int k3 = 1; k3 < KSEL; ++k3) {
                        float v = sW[m][k3];
                        if (v > nmx) { nmx = v; nmp = k3; }
                    }
                    mx = nmx; mp = nmp;
                }
            }
        }
    }
    __syncthreads();

    // ---- write per-split partial result (keys + train indices) ----
    for (int e = t; e < MT * KSEL; e += TB) {
        int m = e >> 5, k = e & 31;
        size_t o = ((size_t)(qBase + m) * NSPLIT + split) * KSEL + k;
        wsW[o] = sW[m][k];
        wsI[o] = sFi[m][k];
    }
}

// --------------- kernel 2: merge splits, weights, vote, argmax --------------
__global__ __launch_bounds__(256)
void knn_final_kernel(const float* __restrict__ wsW,
                      const int*   __restrict__ wsI,
                      const int*   __restrict__ labels,
                      float*       __restrict__ out) {
    __shared__ float sLw[256][KSEL];
    __shared__ int   sLi[256][KSEL];
    int q = blockIdx.x * blockDim.x + threadIdx.x;
    if (q >= QTOT) return;
    int t = threadIdx.x;

    float mx = FINF; int mp = 0;
#pragma unroll
    for (int k = 0; k < KSEL; ++k) { sLw[t][k] = FINF; sLi[t][k] = 0; }

    const float* pw = wsW + (size_t)q * NSPLIT * KSEL;
    const int*   pi = wsI + (size_t)q * NSPLIT * KSEL;
    for (int e = 0; e < NSPLIT * KSEL; ++e) {
        float key = pw[e];
        if (key < mx) {
            sLw[t][mp] = key; sLi[t][mp] = pi[e];
            float nmx = sLw[t][0]; int nmp = 0;
#pragma unroll
            for (int k = 1; k < KSEL; ++k) {
                float v = sLw[t][k];
                if (v > nmx) { nmx = v; nmp = k; }
            }
            mx = nmx; mp = nmp;
        }
    }

    // exact IEEE weights only for the 32 survivors; accumulate class votes
    float votes[NCLS];
#pragma unroll
    for (int c = 0; c < NCLS; ++c) votes[c] = 0.f;
    for (int k = 0; k < KSEL; ++k) {
        float key = sLw[t][k];
        bool  ok  = key < 0.5f * FINF;
        float w   = ok ? (1.0f / key) : 0.f;     // reference: weights = 1/sim
        int   lab = ok ? labels[sLi[t][k]] : -1;
#pragma unroll
        for (int c = 0; c < NCLS; ++c) votes[c] += (lab == c) ? w : 0.f;
    }

    int best = 0; float bv = votes[0];
#pragma unroll
    for (int c = 1; c < NCLS; ++c) {
        if (votes[c] > bv) { bv = votes[c]; best = c; }   // first-max tie rule
    }
    out[q] = (float)best;
}

// ----------------------------- launcher -------------------------------------
extern "C" void kernel_launch(void* const* d_in, const int* in_sizes, int n_in,
                              void* d_out, int out_size, void* d_ws, size_t ws_size,
                              hipStream_t stream) {
    const float* queries = (const float*)d_in[0];   // [1024, 64]   fp32
    const float* train   = (const float*)d_in[1];   // [200000, 64] fp32
    const int*   labels  = (const int*)d_in[2];     // [200000]     int32
    float* out = (float*)d_out;                     // [1024]

    // workspace layout: x2 | partial keys | partial indices (~2.8 MB)
    float* x2  = (float*)d_ws;                                   // NTRAIN
    float* wsW = x2 + NTRAIN;                                    // Q*NSPLIT*K
    int*   wsI = (int*)(wsW + (size_t)QTOT * NSPLIT * KSEL);     // Q*NSPLIT*K

    knn_x2_kernel<<<(NTRAIN + 255) / 256, 256, 0, stream>>>(train, x2, NTRAIN);

    dim3 g1(QTOT / MT, NSPLIT);   // 64 x 8 = 512 workgroups
    knn_partial_kernel<<<g1, TB, 0, stream>>>(queries, train, x2, wsW, wsI);

    knn_final_kernel<<<(QTOT + 255) / 256, 256, 0, stream>>>(wsW, wsI, labels, out);
}